// SmoothnessRegularization_35940286332981
// MI455X (gfx1250) — compile-verified
//
#include <hip/hip_runtime.h>
#include <hip/hip_bf16.h>

typedef __attribute__((ext_vector_type(2))) float v2f;
typedef __attribute__((ext_vector_type(8))) float v8f;

#define LAMBDA_S 0.01f
#define KNBR 5

// Insert (val, idx) into ascending top-5 list held in registers (constant indices only).
__device__ __forceinline__ void sr_insert5(float val, int idx, float bd[5], int bi[5]) {
    if (val < bd[4]) {
        bd[4] = val; bi[4] = idx;
        if (bd[4] < bd[3]) { float t = bd[3]; bd[3] = bd[4]; bd[4] = t; int u = bi[3]; bi[3] = bi[4]; bi[4] = u; }
        if (bd[3] < bd[2]) { float t = bd[2]; bd[2] = bd[3]; bd[3] = t; int u = bi[2]; bi[2] = bi[3]; bi[3] = u; }
        if (bd[2] < bd[1]) { float t = bd[1]; bd[1] = bd[2]; bd[2] = t; int u = bi[1]; bi[1] = bi[2]; bi[2] = u; }
        if (bd[1] < bd[0]) { float t = bd[0]; bd[0] = bd[1]; bd[1] = t; int u = bi[0]; bi[0] = bi[1]; bi[1] = u; }
    }
}

__global__ void sr_init_kernel(float* acc) {
    acc[0] = 0.0f;
    acc[1] = 0.0f;
}

// One wave32 per block; block handles 16 rows. Grid = N/16 blocks.
__global__ __launch_bounds__(32) void sr_knn_smooth_kernel(
    const float* __restrict__ pos,   // [N,3]
    const float* __restrict__ w,     // [N]
    const float* __restrict__ ls,    // [N,3]
    float* __restrict__ acc,         // acc[0]=sum w_diff^2, acc[1]=sum s_diff^2
    int n)
{
    __shared__ float lds_d2[16 * 16];   // 16x16 distance^2 tile (row-major)
    __shared__ float lds_md[32 * 5];    // per-lane candidate dists
    __shared__ int   lds_mi[32 * 5];    // per-lane candidate indices

    const int lane = threadIdx.x;       // 0..31
    const int m0   = blockIdx.x * 16;   // first row of this tile
    const bool lo  = lane < 16;
    const int  c16 = lane & 15;

    // ---- A-matrix registers: 16x4 f32 layout (lanes 0-15: K=0,1 ; lanes 16-31: K=2,3(=0)) ----
    v2f amat;
    {
        int row = m0 + c16;
        float p0 = pos[row * 3 + 0];
        float p1 = pos[row * 3 + 1];
        float p2 = pos[row * 3 + 2];
        amat.x = lo ? p0 : p2;
        amat.y = lo ? p1 : 0.0f;
    }

    // ---- per-lane squared norms of the 8 D-rows this lane produces (M = v + (lo?0:8)) ----
    float smrow[8];
#pragma unroll
    for (int v = 0; v < 8; ++v) {
        int row = m0 + v + (lo ? 0 : 8);
        float p0 = pos[row * 3 + 0];
        float p1 = pos[row * 3 + 1];
        float p2 = pos[row * 3 + 2];
        smrow[v] = p0 * p0 + p1 * p1 + p2 * p2;
    }

    // ---- running top-5 (2 lanes per row: lane handles row lane>>1, cols (lane&1)*8..+7) ----
    float bd[5]; int bi[5];
#pragma unroll
    for (int j = 0; j < 5; ++j) { bd[j] = 3.4e38f; bi[j] = 0; }
    const int selRow = lane >> 1;
    const int selC0  = (lane & 1) * 8;

    const int ntiles = n >> 4;
    for (int nt = 0; nt < ntiles; ++nt) {
        const int n0   = nt * 16;
        const int cidx = n0 + c16;

        // B-matrix registers (4x16 f32 layout == A layout for a Gram matrix),
        // pre-scaled by -2 so the WMMA directly produces -2*G.
        float p0 = pos[cidx * 3 + 0];
        float p1 = pos[cidx * 3 + 1];
        float p2 = pos[cidx * 3 + 2];
        float sqc = p0 * p0 + p1 * p1 + p2 * p2;
        v2f bmat;
        bmat.x = lo ? (-2.0f * p0) : (-2.0f * p2);
        bmat.y = lo ? (-2.0f * p1) : 0.0f;

        // C accumulator preloaded with |m|^2 + |n|^2 so D = |m|^2 + |n|^2 - 2*G = d^2.
        v8f cacc;
#pragma unroll
        for (int v = 0; v < 8; ++v) cacc[v] = smrow[v] + sqc;

        // 8 args: neg_a, A, neg_b, B, c_mod, C, reuse_a, reuse_b
        v8f d = __builtin_amdgcn_wmma_f32_16x16x4_f32(
            false, amat, false, bmat, (short)0, cacc, false, false);

        // spill d^2 tile to LDS to transpose the WMMA lane layout
#pragma unroll
        for (int v = 0; v < 8; ++v) {
            int rr = v + (lo ? 0 : 8);
            lds_d2[rr * 16 + c16] = d[v];
        }
        __syncthreads();

        // selection: each lane scans 8 candidates of its half-row
#pragma unroll
        for (int cc = 0; cc < 8; ++cc) {
            int col  = selC0 + cc;
            float val = lds_d2[selRow * 16 + col];
            int gidx = n0 + col;
            if (gidx != m0 + selRow) {   // exclude self (== reference's idx[:,1:])
                sr_insert5(val, gidx, bd, bi);
            }
        }
        __syncthreads();
    }

    // ---- merge the two half-row lists, then accumulate the regularizer ----
#pragma unroll
    for (int j = 0; j < 5; ++j) { lds_md[lane * 5 + j] = bd[j]; lds_mi[lane * 5 + j] = bi[j]; }
    __syncthreads();

    if (lane < 16) {
        float fd[5]; int fi[5];
#pragma unroll
        for (int j = 0; j < 5; ++j) { fd[j] = 3.4e38f; fi[j] = 0; }
#pragma unroll
        for (int h = 0; h < 2; ++h) {
            int src = lane * 2 + h;
#pragma unroll
            for (int j = 0; j < 5; ++j) {
                sr_insert5(lds_md[src * 5 + j], lds_mi[src * 5 + j], fd, fi);
            }
        }

        int m = m0 + lane;
        float wm = w[m];
        float l0 = ls[m * 3 + 0];
        float l1 = ls[m * 3 + 1];
        float l2 = ls[m * 3 + 2];
        float aw = 0.0f, as = 0.0f;
#pragma unroll
        for (int j = 0; j < 5; ++j) {
            int nb = fi[j];
            float dw = wm - w[nb];
            aw += dw * dw;
            float d0 = l0 - ls[nb * 3 + 0];
            float d1 = l1 - ls[nb * 3 + 1];
            float d2 = l2 - ls[nb * 3 + 2];
            as += d0 * d0 + d1 * d1 + d2 * d2;
        }
        atomicAdd(&acc[0], aw);
        atomicAdd(&acc[1], as);
    }
}

__global__ void sr_finalize_kernel(const float* acc, float* out, int n) {
    float denomW = (float)n * (float)KNBR;          // mean over N*5 w-diffs
    float denomS = (float)n * (float)KNBR * 3.0f;   // mean over N*5*3 s-diffs
    out[0] = LAMBDA_S * (acc[0] / denomW + acc[1] / denomS);
}

extern "C" void kernel_launch(void* const* d_in, const int* in_sizes, int n_in,
                              void* d_out, int out_size, void* d_ws, size_t ws_size,
                              hipStream_t stream) {
    const float* pos = (const float*)d_in[0];   // [N,3]
    const float* w   = (const float*)d_in[1];   // [N]
    const float* ls  = (const float*)d_in[2];   // [N,3]
    float* acc = (float*)d_ws;                  // 2 floats of scratch
    float* out = (float*)d_out;

    const int n = in_sizes[1];                  // N = 16384

    sr_init_kernel<<<1, 1, 0, stream>>>(acc);
    sr_knn_smooth_kernel<<<n / 16, 32, 0, stream>>>(pos, w, ls, acc, n);
    sr_finalize_kernel<<<1, 1, 0, stream>>>(acc, out, n);
}